// VectorKuramoto_33681133536098
// MI455X (gfx1250) — compile-verified
//
#include <hip/hip_runtime.h>
#include <cstddef>

#define NN_ 2048
#define BB_ 4
#define DD_ 4

typedef __attribute__((ext_vector_type(2))) float v2f;
typedef __attribute__((ext_vector_type(8))) float v8f;

// Pass 1: S = sin(theta), C = cos(theta) for all [B,N,D] (128 KB each, in d_ws).
__global__ void kuramoto_sincos_pre(const float* __restrict__ theta,
                                    float* __restrict__ ssin,
                                    float* __restrict__ scos, int n) {
  int idx = blockIdx.x * blockDim.x + threadIdx.x;
  if (idx < n) {
    float s, c;
    __sincosf(theta[idx], &s, &c);
    ssin[idx] = s;
    scos[idx] = c;
  }
}

// Pass 2: streaming pass over affinity/alpha_t/alpha_bias with WMMA j-reduction.
// Block = (i-tile of 16 rows, batch b); 8 wave32s split the j (K) dimension.
// B-tile packs [S | C] into columns 0..7, so one P-wmma and one Q-wmma per
// K-step produce PS|PC and QS|QC; the cross-column combine happens during the
// LDS cross-wave reduction, where it is free.
__global__ __launch_bounds__(256) void kuramoto_wmma_main(
    const float* __restrict__ theta, const float* __restrict__ gamma,
    const float* __restrict__ aff, const float* __restrict__ alpha_t,
    const float* __restrict__ omega, const float* __restrict__ kappa,
    const float* __restrict__ alpha_bias,
    const float* __restrict__ ssin, const float* __restrict__ scos,
    float* __restrict__ out) {
  const int N = NN_, D = DD_;
  const int tile = blockIdx.x;   // 0 .. N/16-1
  const int b = blockIdx.y;      // 0 .. B-1
  const int i0 = tile * 16;
  const int tid = (int)threadIdx.x;
  const int lane = tid & 31;
  const int w = tid >> 5;        // wave id 0..7

  const int m = lane & 15;                 // A: row-in-tile; B/C/D: column
  const bool lo = (lane < 16);
  const int kbase = lo ? 0 : 2;            // K split across wave halves

  const int row = i0 + m;
  const size_t bn_row = ((size_t)b * N + row) * N;  // affinity / alpha_t row
  const size_t ab_row = (size_t)row * N;            // alpha_bias row

  // LDS: 64KB stage of S|C for this batch, overlapped with 16KB reduction
  // buffer (disjoint live ranges, separated by __syncthreads).
  __shared__ union SMem {
    float stage[2 * NN_ * DD_];     // [0..8191]=S[b,:,:], [8192..16383]=C[b,:,:]
    float red[8][2][256];           // [wave][P/Q][16x16 tile]
  } smem;

  // Coalesced b128 staging of S and C for batch b.
  {
    const float4* s4 = (const float4*)(ssin + (size_t)b * N * D);
    const float4* c4 = (const float4*)(scos + (size_t)b * N * D);
    float4* ls = (float4*)(smem.stage);
    float4* lc = (float4*)(smem.stage + N * D);
#pragma unroll
    for (int t = tid; t < (N * D) / 4; t += 256) {
      ls[t] = s4[t];
      lc[t] = c4[t];
    }
  }
  __syncthreads();

  // Per-lane constant part of the B-tile LDS offset: column n=m selects
  // S (n<4, d=n) or C (4<=n<8, d=n-4); n>=8 lanes are masked to zero.
  const int moff = (m & 3) + ((m & 4) ? (N * D) : 0);
  const bool bactive = (m < 8);

  v8f accP = {}, accQ = {};

  const int j_begin = w * (N / 8);
  const int j_end = j_begin + (N / 8);
  for (int j0 = j_begin; j0 < j_end; j0 += 4) {
    const int col = j0 + kbase;
    // float2 covers this lane's two A-VGPR slots (K = kbase, kbase+1)
    v2f av = *(const v2f*)(aff + bn_row + col);
    v2f tv = *(const v2f*)(alpha_t + bn_row + col);
    v2f bv = *(const v2f*)(alpha_bias + ab_row + col);

    // Pull the streaming rows ahead (global_prefetch_b8).
    __builtin_prefetch(aff + bn_row + col + 256, 0, 0);
    __builtin_prefetch(alpha_t + bn_row + col + 256, 0, 0);

    float s0, c0, s1, c1;
    __sincosf(tv.x + bv.x, &s0, &c0);
    __sincosf(tv.y + bv.y, &s1, &c1);

    v2f P, Q;
    P.x = av.x * c0; P.y = av.y * c1;   // aff * cos(alpha_total)
    Q.x = av.x * s0; Q.y = av.y * s1;   // aff * sin(alpha_total)

    // B tile (4x16) = [S | C | 0...]: unconditional ds_load + cndmask.
    const float* bp = smem.stage + (col * 4 + moff);
    const float b0 = bp[0];   // j = col
    const float b1 = bp[4];   // j = col + 1
    v2f Bsc;
    Bsc.x = bactive ? b0 : 0.0f;
    Bsc.y = bactive ? b1 : 0.0f;

    accP = __builtin_amdgcn_wmma_f32_16x16x4_f32(false, P, false, Bsc,
                                                 (short)0, accP, false, false);
    accQ = __builtin_amdgcn_wmma_f32_16x16x4_f32(false, Q, false, Bsc,
                                                 (short)0, accQ, false, false);
  }

  // All waves are done reading the stage region before we overwrite it.
  __syncthreads();

  // C/D layout: VGPR r, lanes 0-15 -> (M=r, NN=lane); lanes 16-31 -> (M=r+8).
#pragma unroll
  for (int r = 0; r < 8; ++r) {
    int e = (r + (lo ? 0 : 8)) * 16 + m;
    smem.red[w][0][e] = accP[r];
    smem.red[w][1][e] = accQ[r];
  }
  __syncthreads();

  if (tid < 64) {
    const int M = tid >> 2;
    const int d = tid & 3;
    const int e = M * 16 + d;
    float ps = 0.f, pc = 0.f, qs = 0.f, qc = 0.f;
#pragma unroll
    for (int ww = 0; ww < 8; ++ww) {
      ps += smem.red[ww][0][e];       // P@S  (cols 0..3)
      pc += smem.red[ww][0][e + 4];   // P@C  (cols 4..7)
      qs += smem.red[ww][1][e];       // Q@S
      qc += smem.red[ww][1][e + 4];   // Q@C
    }
    const int i = i0 + M;
    const size_t idx = ((size_t)b * N + i) * D + d;
    const float th = theta[idx];
    const float ci = scos[idx];   // cos(theta[b,i,d]) from pass 1
    const float si = ssin[idx];
    const float coupling = (1.0f / (float)N) * (ci * (ps - qc) - si * (qs + pc));
    const float drive = kappa[i * D + d] * (gamma[idx] - th);
    out[idx] = th + omega[i * D + d] + drive + coupling;  // DT = 1
  }
}

extern "C" void kernel_launch(void* const* d_in, const int* in_sizes, int n_in,
                              void* d_out, int out_size, void* d_ws, size_t ws_size,
                              hipStream_t stream) {
  const float* theta      = (const float*)d_in[0];
  const float* gamma      = (const float*)d_in[1];
  const float* aff        = (const float*)d_in[2];
  const float* alpha_t    = (const float*)d_in[3];
  const float* omega      = (const float*)d_in[4];
  const float* kappa      = (const float*)d_in[5];
  const float* alpha_bias = (const float*)d_in[6];
  float* out = (float*)d_out;

  float* ssin = (float*)d_ws;                    // [B*N*D]
  float* scos = ssin + (size_t)BB_ * NN_ * DD_;  // [B*N*D]  (total 256 KB)

  const int n = BB_ * NN_ * DD_;
  kuramoto_sincos_pre<<<(n + 255) / 256, 256, 0, stream>>>(theta, ssin, scos, n);

  dim3 grid(NN_ / 16, BB_);
  kuramoto_wmma_main<<<grid, 256, 0, stream>>>(theta, gamma, aff, alpha_t,
                                               omega, kappa, alpha_bias,
                                               ssin, scos, out);
}